// Elongated_Align_90185723281677
// MI455X (gfx1250) — compile-verified
//
#include <hip/hip_runtime.h>
#include <stdint.h>

#define KNN   16
#define BLOCK 256
#define SLAB_ELEMS (BLOCK * KNN)   // 4096 i64 indices = 32 KB per block

typedef uint32_t u32x4 __attribute__((ext_vector_type(4)));
typedef int      i32x4 __attribute__((ext_vector_type(4)));
typedef int      i32x8 __attribute__((ext_vector_type(8)));

// ---------------------------------------------------------------------------
// Kernel 1: one thread per point.
//  - Stage the block's knn_index slab (32 KB) into LDS with the CDNA5 Tensor
//    Data Mover: one tensor_load_to_lds DMA issued by wave 0 (TENSORcnt path),
//    falling back to global_load_async_to_lds_b128 (ASYNCcnt) if the TDM
//    builtin is unavailable.
//  - Per-thread: e from quaternion (norm-free 2/|q|^2 form), 16 gathers
//    (xyz table is 12 MB -> resident in the 192 MB L2), accumulate
//    S|proj|, Sproj^2, S|nbr|^2.
//  - c1 = Sproj^2 - (S|proj|)^2/K   (variance numerator for loss_d)
//    c2 = S|nbr|^2 - Sproj^2        (perp energy for loss_normal, |e|=1)
//  - LDS tree-reduce 256 threads -> per-block partial pair in d_ws.
// ---------------------------------------------------------------------------
__global__ __launch_bounds__(BLOCK) void elong_align_partial(
    const float* __restrict__ xyz,
    const float* __restrict__ rot,
    const long long* __restrict__ knn,
    const int* __restrict__ directp,
    float* __restrict__ partials,
    int N)
{
    __shared__ __align__(16) long long sIdx[SLAB_ELEMS];   // 32 KB index slab
    __shared__ float red1[BLOCK];
    __shared__ float red2[BLOCK];

    const int tid = threadIdx.x;
    const int n   = blockIdx.x * BLOCK + tid;
    const bool fullBlock = ((blockIdx.x + 1) * BLOCK) <= N;   // uniform per block

    if (fullBlock) {
#if __has_builtin(__builtin_amdgcn_tensor_load_to_lds)
        if (tid < 32) {   // wave 0 issues the DMA; TDM ignores EXEC
            const uint64_t ga  = (uint64_t)(uintptr_t)(knn + (size_t)blockIdx.x * SLAB_ELEMS);
            const uint32_t lds = (uint32_t)(uintptr_t)(&sIdx[0]);  // low 32 bits = LDS offset

            // D# group 0: count=1 | lds_addr | global_addr[56:0] | type=2
            const u32x4 g0 = { 1u,
                               lds,
                               (uint32_t)ga,
                               ((uint32_t)(ga >> 32) & 0x01FFFFFFu) | (2u << 30) };
            // D# group 1: mask=0, data_size=3 (8B); tensor_dim0=tile_dim0=4096;
            //             tensor_dim1=tile_dim1=1; tensor_dim0_stride=4096.
            const i32x8 g1 = { 0x00030000,                         // [17:16] data_size=3
                               (int)((SLAB_ELEMS & 0xFFFF) << 16), // [63:48] tensor_dim0 lo
                               0x00010000,                         // [95:80] tensor_dim1=1
                               (int)(SLAB_ELEMS << 16),            // [127:112] tile_dim0=4096
                               1,                                  // [143:128] tile_dim1=1
                               SLAB_ELEMS,                         // [191:160] dim0_stride lo
                               0, 0 };
            const i32x4 g2 = { 0, 0, 0, 0 };   // dims >= 2 unused
            const i32x4 g3 = { 0, 0, 0, 0 };
            const i32x8 g4 = { 0, 0, 0, 0, 0, 0, 0, 0 };   // aux group (6-arg form), unused
            __builtin_amdgcn_tensor_load_to_lds(g0, g1, g2, g3, g4, 0);
            __builtin_amdgcn_s_wait_tensorcnt(0);
        }
#else
        // Fallback: coalesced async copy, 8 rounds x (256 threads x 16 B) = 32 KB.
        const uint64_t gbase = (uint64_t)(uintptr_t)(knn + (size_t)blockIdx.x * SLAB_ELEMS);
        const uint32_t lbase = (uint32_t)(uintptr_t)(&sIdx[0]);
#pragma unroll
        for (int r = 0; r < 8; ++r) {
            const uint32_t byteOff = (uint32_t)(r * 4096 + tid * 16);
            const uint64_t gaddr = gbase + byteOff;
            const uint32_t laddr = lbase + byteOff;
            asm volatile("global_load_async_to_lds_b128 %0, %1, off"
                         :: "v"(laddr), "v"(gaddr) : "memory");
        }
        asm volatile("s_wait_asynccnt 0" ::: "memory");
#endif
    }
    __syncthreads();

    float c1 = 0.0f, c2 = 0.0f;
    if (n < N) {
        // e = column `direct` of rotation matrix; use s = 2/|q|^2 (no sqrt needed).
        const int dcol = *directp;
        const float4 q = ((const float4*)rot)[n];
        const float w = q.x, x = q.y, y = q.z, z = q.w;
        const float s = 2.0f / (w * w + x * x + y * y + z * z);
        float ex, ey, ez;
        if (dcol == 0)      { ex = 1.0f - s * (y * y + z * z); ey = s * (x * y + w * z); ez = s * (x * z - w * y); }
        else if (dcol == 1) { ex = s * (x * y - w * z); ey = 1.0f - s * (x * x + z * z); ez = s * (y * z + w * x); }
        else                { ex = s * (x * z + w * y); ey = s * (y * z - w * x); ez = 1.0f - s * (x * x + y * y); }

        const float* cp = xyz + 3 * (size_t)n;
        const float cx = cp[0], cy = cp[1], cz = cp[2];

        float sa = 0.0f, sp2 = 0.0f, sn2 = 0.0f;
        auto acc = [&](long long id) {
            const float* p = xyz + 3 * (size_t)id;      // gather: hot in 192 MB L2
            const float dx = p[0] - cx, dy = p[1] - cy, dz = p[2] - cz;
            const float pr = fmaf(dx, ex, fmaf(dy, ey, dz * ez));
            sa += fabsf(pr);
            sp2 = fmaf(pr, pr, sp2);
            sn2 = fmaf(dx, dx, fmaf(dy, dy, fmaf(dz, dz, sn2)));
        };

        if (fullBlock) {
            // Thread's 16 indices are 128 contiguous bytes in LDS (stride 128 B per
            // thread -> 16-way bank conflict if read in-order). Rotate the 16 B chunk
            // by tid so concurrent lanes hit different banks (<= 2-way).
            const longlong2* row = (const longlong2*)(&sIdx[tid * KNN]);
#pragma unroll
            for (int j = 0; j < 8; ++j) {
                const longlong2 pr2 = row[(j + tid) & 7];   // ds_load_b128
                acc(pr2.x);
                acc(pr2.y);
            }
        } else {
            const long long* row = knn + (size_t)n * KNN;
#pragma unroll
            for (int k = 0; k < KNN; ++k) acc(row[k]);
        }

        c1 = sp2 - sa * sa * (1.0f / (float)KNN);
        c2 = sn2 - sp2;
    }

    red1[tid] = c1;
    red2[tid] = c2;
    __syncthreads();
#pragma unroll
    for (int sh = BLOCK / 2; sh > 0; sh >>= 1) {
        if (tid < sh) { red1[tid] += red1[tid + sh]; red2[tid] += red2[tid + sh]; }
        __syncthreads();
    }
    if (tid == 0) {
        partials[2 * blockIdx.x + 0] = red1[0];
        partials[2 * blockIdx.x + 1] = red2[0];
    }
}

// ---------------------------------------------------------------------------
// Kernel 2: single block reduces the per-block partials (double accumulation)
// and writes the two scalar losses, scaled by 1/(N*K). No atomics on d_out, so
// the poisoned output buffer is fully overwritten every call (deterministic).
// ---------------------------------------------------------------------------
__global__ __launch_bounds__(256) void elong_align_final(
    const float* __restrict__ partials, int nblocks,
    float* __restrict__ out, double scale)
{
    __shared__ double r1[256];
    __shared__ double r2[256];
    double a = 0.0, b = 0.0;
    for (int i = threadIdx.x; i < nblocks; i += 256) {
        a += (double)partials[2 * i + 0];
        b += (double)partials[2 * i + 1];
    }
    r1[threadIdx.x] = a;
    r2[threadIdx.x] = b;
    __syncthreads();
#pragma unroll
    for (int sh = 128; sh > 0; sh >>= 1) {
        if (threadIdx.x < sh) { r1[threadIdx.x] += r1[threadIdx.x + sh]; r2[threadIdx.x] += r2[threadIdx.x + sh]; }
        __syncthreads();
    }
    if (threadIdx.x == 0) {
        out[0] = (float)(r1[0] * scale);   // loss_d
        out[1] = (float)(r2[0] * scale);   // loss_normal
    }
}

extern "C" void kernel_launch(void* const* d_in, const int* in_sizes, int n_in,
                              void* d_out, int out_size, void* d_ws, size_t ws_size,
                              hipStream_t stream)
{
    const float*     xyz    = (const float*)d_in[0];
    /* d_in[1] = xyz_id : unused by the reference computation */
    const float*     rot    = (const float*)d_in[2];
    const long long* knn    = (const long long*)d_in[3];
    const int*       direct = (const int*)d_in[4];
    float*           out    = (float*)d_out;

    const int N = in_sizes[0] / 3;                 // xyz is (N,3)
    const int K = KNN;                             // reference fixes K = 16
    const int nblocks = (N + BLOCK - 1) / BLOCK;

    float* partials = (float*)d_ws;                // 2 floats per block

    elong_align_partial<<<nblocks, BLOCK, 0, stream>>>(xyz, rot, knn, direct, partials, N);

    const double scale = 1.0 / ((double)N * (double)K);
    elong_align_final<<<1, 256, 0, stream>>>(partials, nblocks, out, scale);
}